// GATLayer_42889543418167
// MI455X (gfx1250) — compile-verified
//
#include <hip/hip_runtime.h>
#include <hip/hip_bf16.h>

typedef float v2f __attribute__((ext_vector_type(2)));
typedef float v8f __attribute__((ext_vector_type(8)));

// ---------------------------------------------------------------------------
// z = h @ W  ( [N,128] x [128,128] fp32 ), via V_WMMA_F32_16X16X4_F32.
// Block = 256 threads = 8 waves. Block -> one 16-row M tile; wave w -> 16-col
// N tile w. K loop: 32 steps of K=4.
// A (16x4) lane layout: lanes 0-15 hold K=k,k+1 of row M=lane;
//                       lanes 16-31 hold K=k+2,k+3 of row M=lane-16.
// B (4x16) lane layout mirrors the K split: lanes 0-15 hold rows k,k+1 at
//                       col n0+lane; lanes 16-31 hold rows k+2,k+3.
// C/D (16x16 f32): VGPR i = row m0 + 8*half + i, col n0 + (lane&15).
// ---------------------------------------------------------------------------
__global__ __launch_bounds__(256) void gat_gemm_z(const float* __restrict__ h,
                                                  const float* __restrict__ W,
                                                  float* __restrict__ z,
                                                  int N) {
  const int wave = threadIdx.x >> 5;       // 0..7 -> N tile
  const int lane = threadIdx.x & 31;
  const int m0   = blockIdx.x * 16;
  const int n0   = wave * 16;
  const int half = lane >> 4;              // 0: lanes 0-15, 1: lanes 16-31
  const int l    = lane & 15;

  int arow = m0 + l;
  if (arow >= N) arow = N - 1;             // clamp loads (select, no EXEC change)

  const float* __restrict__ hrow = h + (size_t)arow * 128 + (half ? 2 : 0);
  const float* __restrict__ wcol = W + (size_t)(half ? 2 : 0) * 128 + n0 + l;

  v8f c = {0.f, 0.f, 0.f, 0.f, 0.f, 0.f, 0.f, 0.f};
#pragma unroll 8
  for (int k = 0; k < 128; k += 4) {
    v2f a, b;
    a.x = hrow[k];
    a.y = hrow[k + 1];
    b.x = wcol[(size_t)k * 128];
    b.y = wcol[(size_t)(k + 1) * 128];
    c = __builtin_amdgcn_wmma_f32_16x16x4_f32(false, a, false, b,
                                              (short)0, c, false, false);
  }

  const int orow = m0 + half * 8;
  float* __restrict__ zout = z + (size_t)orow * 128 + n0 + l;
  if (m0 + 16 <= N) {
    // Full tile (every block except possibly the last): unguarded stores.
#pragma unroll
    for (int i = 0; i < 8; ++i) zout[(size_t)i * 128] = c[i];
  } else {
    // Tail tile only: per-row guard.
#pragma unroll
    for (int i = 0; i < 8; ++i) {
      if (orow + i < N) zout[(size_t)i * 128] = c[i];
    }
  }
}

// ---------------------------------------------------------------------------
// el[n] = dot(z[n], w_l), er[n] = dot(z[n], w_r). One wave32 per node,
// float4 slice per lane + shfl-xor reduction.
// ---------------------------------------------------------------------------
__global__ __launch_bounds__(256) void gat_attn_proj(const float* __restrict__ z,
                                                     const float* __restrict__ wl,
                                                     const float* __restrict__ wr,
                                                     float* __restrict__ el,
                                                     float* __restrict__ er,
                                                     int N) {
  const int node = blockIdx.x * 8 + (threadIdx.x >> 5);
  const int lane = threadIdx.x & 31;
  if (node >= N) return;

  const float4 zv = ((const float4*)(z + (size_t)node * 128))[lane];
  const float4 lv = ((const float4*)wl)[lane];
  const float4 rv = ((const float4*)wr)[lane];

  float pl = zv.x * lv.x + zv.y * lv.y + zv.z * lv.z + zv.w * lv.w;
  float pr = zv.x * rv.x + zv.y * rv.y + zv.z * rv.z + zv.w * rv.w;
#pragma unroll
  for (int off = 16; off > 0; off >>= 1) {
    pl += __shfl_xor(pl, off, 32);
    pr += __shfl_xor(pr, off, 32);
  }
  if (lane == 0) {
    el[node] = pl;
    er[node] = pr;
  }
}

// ---------------------------------------------------------------------------
// e[i] = leaky_relu(el[src[i]] + er[dst[i]], 0.01)
// ---------------------------------------------------------------------------
__global__ __launch_bounds__(256) void gat_edge_score(const int* __restrict__ src,
                                                      const int* __restrict__ dst,
                                                      const float* __restrict__ el,
                                                      const float* __restrict__ er,
                                                      float* __restrict__ e,
                                                      int E) {
  const int i = blockIdx.x * 256 + threadIdx.x;
  if (i >= E) return;
  const float v = el[src[i]] + er[dst[i]];
  e[i] = v > 0.f ? v : 0.01f * v;
}

// ---------------------------------------------------------------------------
// rp[n] = lower_bound(dst, n): dst is sorted, so edges for node n live in
// [rp[n], rp[n+1]).
// ---------------------------------------------------------------------------
__global__ __launch_bounds__(256) void gat_rowptr(const int* __restrict__ dst,
                                                  int* __restrict__ rp,
                                                  int N, int E) {
  const int n = blockIdx.x * 256 + threadIdx.x;
  if (n > N) return;
  int lo = 0, hi = E;
  while (lo < hi) {
    const int mid = (lo + hi) >> 1;
    if (dst[mid] < n) lo = mid + 1; else hi = mid;
  }
  rp[n] = lo;
}

// ---------------------------------------------------------------------------
// Per-dst segment softmax + weighted gather-sum of z[src]. One wave32 per
// node: max pass, sum-exp pass (shfl reductions), then serial edge loop with
// each lane owning a float4 column slice. Empty segment -> zeros.
// ---------------------------------------------------------------------------
__global__ __launch_bounds__(256) void gat_aggregate(const int* __restrict__ src,
                                                     const float* __restrict__ e,
                                                     const float* __restrict__ z,
                                                     const int* __restrict__ rp,
                                                     float* __restrict__ out,
                                                     int N) {
  const int node = blockIdx.x * 8 + (threadIdx.x >> 5);
  const int lane = threadIdx.x & 31;
  if (node >= N) return;

  const int r0 = rp[node];
  const int r1 = rp[node + 1];

  float4 acc = make_float4(0.f, 0.f, 0.f, 0.f);
  if (r0 < r1) {
    float m = -3.402823466e38f;
    for (int j = r0 + lane; j < r1; j += 32) m = fmaxf(m, e[j]);
#pragma unroll
    for (int off = 16; off > 0; off >>= 1) m = fmaxf(m, __shfl_xor(m, off, 32));

    float s = 0.f;
    for (int j = r0 + lane; j < r1; j += 32) s += __expf(e[j] - m);
#pragma unroll
    for (int off = 16; off > 0; off >>= 1) s += __shfl_xor(s, off, 32);
    const float inv = 1.f / s;

    for (int j = r0; j < r1; ++j) {
      const float w = __expf(e[j] - m) * inv;                 // uniform in wave
      const float4 zv = ((const float4*)(z + (size_t)src[j] * 128))[lane];
      acc.x += w * zv.x;
      acc.y += w * zv.y;
      acc.z += w * zv.z;
      acc.w += w * zv.w;
    }
  }
  ((float4*)(out + (size_t)node * 128))[lane] = acc;
}

extern "C" void kernel_launch(void* const* d_in, const int* in_sizes, int n_in,
                              void* d_out, int out_size, void* d_ws, size_t ws_size,
                              hipStream_t stream) {
  const float* h   = (const float*)d_in[0];
  const int*   src = (const int*)d_in[1];
  const int*   dst = (const int*)d_in[2];
  const float* W   = (const float*)d_in[3];
  const float* wl  = (const float*)d_in[4];
  const float* wr  = (const float*)d_in[5];
  float*       out = (float*)d_out;

  const int N = in_sizes[0] / 128;
  const int E = in_sizes[1];

  char* ws = (char*)d_ws;
  float* z  = (float*)ws;  ws += (size_t)N * 128 * sizeof(float);
  float* el = (float*)ws;  ws += (size_t)N * sizeof(float);
  float* er = (float*)ws;  ws += (size_t)N * sizeof(float);
  float* e  = (float*)ws;  ws += (size_t)E * sizeof(float);
  int*   rp = (int*)ws;

  gat_gemm_z<<<(N + 15) / 16, 256, 0, stream>>>(h, W, z, N);
  gat_attn_proj<<<(N + 7) / 8, 256, 0, stream>>>(z, wl, wr, el, er, N);
  gat_edge_score<<<(E + 255) / 256, 256, 0, stream>>>(src, dst, el, er, e, E);
  gat_rowptr<<<(N + 256) / 256, 256, 0, stream>>>(dst, rp, N, E);
  gat_aggregate<<<(N + 7) / 8, 256, 0, stream>>>(src, e, z, rp, out, N);
}